// OurModel_64106681860754
// MI455X (gfx1250) — compile-verified
//
#include <hip/hip_runtime.h>
#include <hip/hip_bf16.h>
#include <math.h>

typedef __attribute__((ext_vector_type(2))) float v2f;
typedef __attribute__((ext_vector_type(8))) float v8f;

#define NN 50001          // NUM_NODES + 1
#define EDGES 800000
#define BV 320            // B*V
#define NTOK 32           // N nodes per visit
#define NEG_SLOPE 0.2f

// =====================================================================
// Generic fp32 WMMA GEMM:  C[M,128] = A[M,128] @ W[128,128] (+bias)(relu)(+resid)
// 256 threads = 8 waves; block computes a 32x128 tile.
// Wave w: rows [16*(w>>2), +16), col pair [32*(w&3), +32) -> TWO independent
// v_wmma_f32_16x16x4_f32 accumulator chains (hides WMMA->WMMA RAW latency,
// reuses each A fragment for two WMMAs).
// =====================================================================
__global__ __launch_bounds__(256) void gemm128(const float* __restrict__ A,
                                               const float* __restrict__ W,
                                               const float* __restrict__ bias,
                                               const float* __restrict__ resid,
                                               float* __restrict__ C,
                                               int M, int relu) {
  __shared__ float As[32 * 128];   // 16 KB
  __shared__ float Ws[64 * 128];   // 32 KB (K staged in 2 chunks)
  const int t  = threadIdx.x;
  const int m0 = blockIdx.x * 32;

  // stage A tile (32 rows), zero-padded past M
  for (int idx = t; idx < 32 * 32; idx += 256) {
    int row = idx >> 5, c4 = idx & 31;
    float4 val = make_float4(0.f, 0.f, 0.f, 0.f);
    if (m0 + row < M)
      val = ((const float4*)(A + (size_t)(m0 + row) * 128))[c4];
    ((float4*)As)[idx] = val;
  }

  const int lane = t & 31;
  const int wave = t >> 5;
  const int r    = lane & 15;      // row (A) / col (B) within a 16x16 tile
  const int hi   = lane >> 4;      // lane-half selects K pair
  const int rh   = wave >> 2;      // row-half of the 32-row block tile
  const int n0   = (wave & 3) * 32;
  const int arow = rh * 16 + r;
  v8f acc0 = {}, acc1 = {};

  for (int kc = 0; kc < 128; kc += 64) {
    __syncthreads();
    for (int idx = t; idx < 64 * 32; idx += 256)
      ((float4*)Ws)[idx] = ((const float4*)(W + (size_t)kc * 128))[idx];
    __syncthreads();
#pragma unroll
    for (int k0 = 0; k0 < 64; k0 += 4) {
      const int kk = k0 + 2 * hi;
      v2f a, b0, b1;
      // A 16x4 frag: VGPR0/1 = K = 2*hi { +0, +1 }, M = r  (ISA 7.12.2)
      a.x  = As[arow * 128 + kc + kk];
      a.y  = As[arow * 128 + kc + kk + 1];
      // B 4x16 frags for the two column tiles
      b0.x = Ws[kk * 128 + n0 + r];
      b0.y = Ws[(kk + 1) * 128 + n0 + r];
      b1.x = Ws[kk * 128 + n0 + 16 + r];
      b1.y = Ws[(kk + 1) * 128 + n0 + 16 + r];
      acc0 = __builtin_amdgcn_wmma_f32_16x16x4_f32(false, a, false, b0,
                                                   (short)0, acc0, false, false);
      acc1 = __builtin_amdgcn_wmma_f32_16x16x4_f32(false, a, false, b1,
                                                   (short)0, acc1, false, false);
    }
  }
  // C/D layout: VGPR j -> row j + 8*hi, col n0 + r
#pragma unroll
  for (int j = 0; j < 8; ++j) {
    int row = m0 + rh * 16 + j + 8 * hi;
    if (row < M) {
      int col0 = n0 + r, col1 = n0 + 16 + r;
      float v0 = acc0[j], v1 = acc1[j];
      if (bias)  { v0 += bias[col0];  v1 += bias[col1]; }
      if (relu)  { v0 = fmaxf(v0, 0.f); v1 = fmaxf(v1, 0.f); }
      if (resid) { v0 += resid[(size_t)row * 128 + col0];
                   v1 += resid[(size_t)row * 128 + col1]; }
      C[(size_t)row * 128 + col0] = v0;
      C[(size_t)row * 128 + col1] = v1;
    }
  }
}

// =====================================================================
// Multihead attention for one set (block = one batch element, wave = one head)
// oq[b,i,:] = q[b,i,:] + softmax(q kT / sqrt(128)) v      (HEADS=4, dh=32)
// =====================================================================
__global__ __launch_bounds__(128) void attn_kernel(
    const float* __restrict__ qg, long long qs,
    const float* __restrict__ kg, long long ks,
    const float* __restrict__ vg, long long vs,
    float* __restrict__ og, long long os,
    int nq, int nk) {
  __shared__ float Qs[32 * 128], Ks[32 * 128], Vs[32 * 128];
  const int b = blockIdx.x;
  const int t = threadIdx.x;
  const float* qb = qg + (size_t)b * qs;
  const float* kb = kg + (size_t)b * ks;
  const float* vb = vg + (size_t)b * vs;
  for (int idx = t; idx < nk * 128; idx += 128) { Ks[idx] = kb[idx]; Vs[idx] = vb[idx]; }
  for (int idx = t; idx < nq * 128; idx += 128) { Qs[idx] = qb[idx]; }
  __syncthreads();
  const int head = t >> 5, lane = t & 31;
  const float scale = 0.08838834764831845f;  // 1/sqrt(128)
  for (int i = 0; i < nq; ++i) {
    float logit = -INFINITY;
    if (lane < nk) {
      float s = 0.f;
#pragma unroll
      for (int d = 0; d < 32; ++d)
        s += Qs[i * 128 + head * 32 + d] * Ks[lane * 128 + head * 32 + d];
      logit = s * scale;
    }
    float mx = logit;
    for (int off = 16; off; off >>= 1) mx = fmaxf(mx, __shfl_xor(mx, off, 32));
    float w = (lane < nk) ? __expf(logit - mx) : 0.f;
    float sum = w;
    for (int off = 16; off; off >>= 1) sum += __shfl_xor(sum, off, 32);
    float a = w / sum;
    float o = 0.f;
    for (int kk = 0; kk < nk; ++kk) {
      float ak = __shfl(a, kk, 32);
      o += ak * Vs[kk * 128 + head * 32 + lane];
    }
    og[(size_t)b * os + i * 128 + head * 32 + lane] =
        Qs[i * 128 + head * 32 + lane] + o;
  }
}

// =====================================================================
// GAT helper kernels
// =====================================================================
__global__ void fill_u32_k(unsigned* p, unsigned v, size_t n) {
  size_t i = (size_t)blockIdx.x * blockDim.x + threadIdx.x;
  if (i < n) p[i] = v;
}

// per-node dot products: as = h . a_src, ad = h . a_dst  (one wave per node)
__global__ void dots_k(const float* __restrict__ h, const float* __restrict__ asrc,
                       const float* __restrict__ adst, float* as_, float* ad_, int n) {
  int wid  = (int)(((size_t)blockIdx.x * blockDim.x + threadIdx.x) >> 5);
  int lane = threadIdx.x & 31;
  if (wid >= n) return;
  const float* row = h + (size_t)wid * 128;
  float s1 = 0.f, s2 = 0.f;
  for (int c = lane; c < 128; c += 32) { float v = row[c]; s1 += v * asrc[c]; s2 += v * adst[c]; }
  for (int off = 16; off; off >>= 1) { s1 += __shfl_xor(s1, off, 32); s2 += __shfl_xor(s2, off, 32); }
  if (lane == 0) { as_[wid] = s1; ad_[wid] = s2; }
}

__device__ inline float leaky(float e) { return e > 0.f ? e : NEG_SLOPE * e; }
__device__ inline unsigned fmap(float f) {
  unsigned u = __float_as_uint(f);
  return (u & 0x80000000u) ? ~u : (u | 0x80000000u);
}

__global__ void edge_max_k(const int* __restrict__ src, const int* __restrict__ dst,
                           const float* __restrict__ as_, const float* __restrict__ ad_,
                           unsigned* __restrict__ mmax) {
  int i = blockIdx.x * blockDim.x + threadIdx.x;
  if (i >= EDGES) return;
  float e = leaky(as_[src[i]] + ad_[dst[i]]);
  atomicMax(mmax + dst[i], fmap(e));
}

__global__ void finalize_m_k(const unsigned* __restrict__ mmax, float* __restrict__ mf, int n) {
  int i = blockIdx.x * blockDim.x + threadIdx.x;
  if (i >= n) return;
  unsigned u = mmax[i];
  float m = 0.f;  // no incoming edges -> isfinite() guard -> 0
  if (u != 0u)
    m = (u & 0x80000000u) ? __uint_as_float(u & 0x7FFFFFFFu) : __uint_as_float(~u);
  mf[i] = m;
}

__global__ void edge_exp_k(const int* __restrict__ src, const int* __restrict__ dst,
                           const float* __restrict__ as_, const float* __restrict__ ad_,
                           const float* __restrict__ mf, float* __restrict__ wexp,
                           float* __restrict__ denom) {
  int i = blockIdx.x * blockDim.x + threadIdx.x;
  if (i >= EDGES) return;
  int d = dst[i];
  float e = leaky(as_[src[i]] + ad_[d]);
  float w = __expf(e - mf[d]);
  wexp[i] = w;
  atomicAdd(denom + d, w);
}

// one wave per edge, 32 lanes x 4 channels of atomic scatter-add
__global__ void edge_aggr_k(const int* __restrict__ src, const int* __restrict__ dst,
                            const float* __restrict__ wexp, const float* __restrict__ denom,
                            const float* __restrict__ h, float* __restrict__ out) {
  int wid  = (int)(((size_t)blockIdx.x * blockDim.x + threadIdx.x) >> 5);
  int lane = threadIdx.x & 31;
  if (wid >= EDGES) return;
  int s = src[wid], d = dst[wid];
  float alpha = wexp[wid] / (denom[d] + 1e-16f);
  const float* hs = h + (size_t)s * 128;
  float* od = out + (size_t)d * 128;
#pragma unroll
  for (int c = 0; c < 4; ++c) {
    int col = lane + 32 * c;
    atomicAdd(od + col, alpha * hs[col]);
  }
}

__global__ void gather_k(const int* __restrict__ ids, const float* __restrict__ aggr,
                         const float* __restrict__ gb, float* __restrict__ X, size_t n) {
  size_t i = (size_t)blockIdx.x * blockDim.x + threadIdx.x;
  if (i >= n) return;
  size_t row = i >> 7;
  int c = (int)(i & 127);
  X[i] = aggr[(size_t)ids[row] * 128 + c] + gb[c];
}

__global__ void final_add_k(const float* __restrict__ X, const float* __restrict__ oe,
                            const float* __restrict__ te, const int* __restrict__ vo,
                            const int* __restrict__ vt, float* __restrict__ out) {
  int i = blockIdx.x * blockDim.x + threadIdx.x;
  if (i >= BV * 128) return;
  int bv = i >> 7, c = i & 127;
  out[i] = X[i] + oe[(size_t)vo[bv] * 128 + c] + te[(size_t)vt[bv] * 128 + c];
}

// =====================================================================
// Host orchestration
// =====================================================================
struct MabP { const float *Wq, *bq, *Wk, *bk, *Wv, *bv, *Wo, *bo; };
static inline MabP mab_at(void* const* d, int i) {
  return MabP{(const float*)d[i],     (const float*)d[i + 1],
              (const float*)d[i + 2], (const float*)d[i + 3],
              (const float*)d[i + 4], (const float*)d[i + 5],
              (const float*)d[i + 6], (const float*)d[i + 7]};
}

extern "C" void kernel_launch(void* const* d_in, const int* in_sizes, int n_in,
                              void* d_out, int out_size, void* d_ws, size_t ws_size,
                              hipStream_t stream) {
  (void)in_sizes; (void)n_in; (void)out_size; (void)ws_size;
  // ---- input map: recursive insertion-order flatten of setup_inputs() ----
  const float* node_embed = (const float*)d_in[0];
  const float* lin_W = (const float*)d_in[1];
  const float* lin_b = (const float*)d_in[2];
  const float* gat_W = (const float*)d_in[3];
  const float* a_src = (const float*)d_in[4];
  const float* a_dst = (const float*)d_in[5];
  const float* gat_b = (const float*)d_in[6];
  const float* I0    = (const float*)d_in[7];
  MabP e0m0 = mab_at(d_in, 8);
  MabP e0m1 = mab_at(d_in, 16);
  const float* I1    = (const float*)d_in[24];
  MabP e1m0 = mab_at(d_in, 25);
  MabP e1m1 = mab_at(d_in, 33);
  const float* pmaS  = (const float*)d_in[41];
  MabP pmaM  = mab_at(d_in, 42);
  MabP sab0  = mab_at(d_in, 50);
  MabP sab1  = mab_at(d_in, 58);
  const float* proj_W = (const float*)d_in[66];
  const float* proj_b = (const float*)d_in[67];
  MabP protoM = mab_at(d_in, 68);
  const float* ffn_W = (const float*)d_in[76];
  const float* ffn_b = (const float*)d_in[77];
  const float* protos = (const float*)d_in[78];   // [5,128]
  const float* order_embed = (const float*)d_in[79];
  const float* time_embed  = (const float*)d_in[80];
  const int* node_ids = (const int*)d_in[81];
  const int* e_src = (const int*)d_in[82];
  const int* e_dst = e_src + EDGES;
  // d_in[83] edge_attr: unused (edge_dim=None in reference)
  const int* visit_times = (const int*)d_in[84];
  const int* visit_order = (const int*)d_in[85];

  // ---- workspace carve (floats, 256B-aligned chunks) ----
  float* w = (float*)d_ws;
  size_t off = 0;
  auto take = [&](size_t n) { float* p = w + off; off += (n + 63) & ~(size_t)63; return p; };
  float* buf0  = take((size_t)NN * 128);   // x, then GAT aggregation output
  float* buf1  = take((size_t)NN * 128);   // h
  float* as_   = take(NN);
  float* ad_   = take(NN);
  unsigned* mmax = (unsigned*)take(NN);
  float* mf    = take(NN);
  float* denom = take(NN);
  float* wexp  = take(EDGES);
  float* Xbuf  = take((size_t)BV * NTOK * 128);
  float* Hbuf  = take((size_t)BV * NTOK * 128);
  float* qbuf  = take((size_t)BV * NTOK * 128);
  float* kbuf  = take((size_t)BV * NTOK * 128);
  float* vbuf  = take((size_t)BV * NTOK * 128);
  float* oqbuf = take((size_t)BV * NTOK * 128);
  float* qI    = take((size_t)NTOK * 128);
  float* X1a   = take((size_t)BV * 128);
  float* X1b   = take((size_t)BV * 128);

  auto gemm = [&](const float* A, const float* Wm, const float* bias,
                  const float* resid, float* C, int M, int relu) {
    gemm128<<<(M + 31) / 32, 256, 0, stream>>>(A, Wm, bias, resid, C, M, relu);
  };
  // MAB: q/k/v proj -> attention(+q residual) -> o + relu(o@Wo+bo)
  auto run_mab = [&](const MabP& p, const float* Qin, int Mq, float* qb,
                     long long qstr, int nq, const float* Kin, int Mk,
                     long long kstr, int nk, float* outb) {
    gemm(Qin, p.Wq, p.bq, nullptr, qb, Mq, 0);
    gemm(Kin, p.Wk, p.bk, nullptr, kbuf, Mk, 0);
    gemm(Kin, p.Wv, p.bv, nullptr, vbuf, Mk, 0);
    attn_kernel<<<BV, 128, 0, stream>>>(qb, qstr, kbuf, kstr, vbuf, kstr,
                                        oqbuf, (long long)nq * 128, nq, nk);
    gemm(oqbuf, p.Wo, p.bo, oqbuf, outb, BV * nq, 1);
  };

  // ================= Phase A: node features + GATConv =================
  gemm(node_embed, lin_W, lin_b, nullptr, buf0, NN, 0);   // x = embed@linW+b
  gemm(buf0, gat_W, nullptr, nullptr, buf1, NN, 0);       // h = x@W
  dots_k<<<((size_t)NN * 32 + 255) / 256, 256, 0, stream>>>(buf1, a_src, a_dst, as_, ad_, NN);
  fill_u32_k<<<(NN + 255) / 256, 256, 0, stream>>>(mmax, 0u, NN);
  fill_u32_k<<<(NN + 255) / 256, 256, 0, stream>>>((unsigned*)denom, 0u, NN);
  edge_max_k<<<(EDGES + 255) / 256, 256, 0, stream>>>(e_src, e_dst, as_, ad_, mmax);
  finalize_m_k<<<(NN + 255) / 256, 256, 0, stream>>>(mmax, mf, NN);
  edge_exp_k<<<(EDGES + 255) / 256, 256, 0, stream>>>(e_src, e_dst, as_, ad_, mf, wexp, denom);
  fill_u32_k<<<((size_t)NN * 128 + 255) / 256, 256, 0, stream>>>((unsigned*)buf0, 0u, (size_t)NN * 128);
  edge_aggr_k<<<((size_t)EDGES * 32 + 255) / 256, 256, 0, stream>>>(e_src, e_dst, wexp, denom, buf1, buf0);
  gather_k<<<((size_t)BV * NTOK * 128 + 255) / 256, 256, 0, stream>>>(
      node_ids, buf0, gat_b, Xbuf, (size_t)BV * NTOK * 128);

  // ================= Phase B: Set Transformer =================
  const long long STR = (long long)NTOK * 128;
  // ISAB 0: H = MAB(I, X) [q broadcast]; X = MAB(X, H)
  run_mab(e0m0, I0, NTOK, qI, 0, NTOK, Xbuf, BV * NTOK, STR, NTOK, Hbuf);
  run_mab(e0m1, Xbuf, BV * NTOK, qbuf, STR, NTOK, Hbuf, BV * NTOK, STR, NTOK, Xbuf);
  // ISAB 1
  run_mab(e1m0, I1, NTOK, qI, 0, NTOK, Xbuf, BV * NTOK, STR, NTOK, Hbuf);
  run_mab(e1m1, Xbuf, BV * NTOK, qbuf, STR, NTOK, Hbuf, BV * NTOK, STR, NTOK, Xbuf);
  // PMA: X1 = MAB(S, X) [q broadcast, nq=1]
  run_mab(pmaM, pmaS, 1, qI, 0, 1, Xbuf, BV * NTOK, STR, NTOK, X1a);
  // SAB x2 (nq=nk=1)
  run_mab(sab0, X1a, BV, qbuf, 128, 1, X1a, BV, 128, 1, X1b);
  run_mab(sab1, X1b, BV, qbuf, 128, 1, X1b, BV, 128, 1, X1a);
  // proj
  gemm(X1a, proj_W, proj_b, nullptr, X1b, BV, 0);
  // prototype MAB: K = 5 shared prototypes (k/v broadcast)
  run_mab(protoM, X1b, BV, qbuf, 128, 1, protos, 5, 0, 5, X1a);
  // ffn
  gemm(X1a, ffn_W, ffn_b, nullptr, X1b, BV, 0);
  // final: + order/time embeddings
  final_add_k<<<(BV * 128 + 255) / 256, 256, 0, stream>>>(
      X1b, order_embed, time_embed, visit_order, visit_times, (float*)d_out);
}